// AdaAttNStar_65515431133290
// MI455X (gfx1250) — compile-verified
//
#include <hip/hip_runtime.h>
#include <math.h>

typedef __attribute__((ext_vector_type(2))) float v2f;
typedef __attribute__((ext_vector_type(8))) float v8f;

#define B_   2
#define C_   3
#define HW_  9216              // 96*96
#define CNT_ (B_ * HW_)        // per-channel reduction count (mean over batch+spatial)
#define EPS_ 1e-12f

// workspace layout (floats):
//   [0..17]  : stats, (t*3+c)*2 -> {sum, sumsq}; t: 0=F_c, 1=F_c_prev, 2=F_s_prev
//   [32..67] : Gram, b*18 + m*6 + n ; n<3: G[m][n]=Σ K[m]V[n], n>=3: G2[m][n-3]=Σ K[m]V[n-3]^2
#define WS_STATS 0
#define WS_GRAM  32

__device__ __forceinline__ void ch_stats(const float* __restrict__ ws, int blk,
                                         float& mean, float& rstd) {
  float s = ws[WS_STATS + blk * 2 + 0];
  float q = ws[WS_STATS + blk * 2 + 1];
  mean = s / (float)CNT_;
  float var = (q - s * mean) / (float)(CNT_ - 1);   // unbiased (ddof=1), matches torch.std
  rstd = 1.0f / (sqrtf(var) + EPS_);
}

// ---------- kernel 1: per-channel sum / sum-of-squares over (batch, H, W) ----------
__global__ void stats_kernel(const float* __restrict__ Fc,
                             const float* __restrict__ Fcp,
                             const float* __restrict__ Fsp,
                             float* __restrict__ ws) {
  __shared__ float s1[256];
  __shared__ float s2[256];
  const int blk = blockIdx.x;            // 0..8
  const int t = blk / 3, c = blk % 3;
  const float* src = (t == 0) ? Fc : (t == 1) ? Fcp : Fsp;
  float a = 0.f, q = 0.f;
  for (int i = threadIdx.x; i < CNT_; i += 256) {
    int b = i / HW_;
    int n = i - b * HW_;
    float x = src[(b * C_ + c) * HW_ + n];
    a += x;
    q += x * x;
  }
  s1[threadIdx.x] = a;
  s2[threadIdx.x] = q;
  __syncthreads();
  for (int o = 128; o > 0; o >>= 1) {
    if ((int)threadIdx.x < o) {
      s1[threadIdx.x] += s1[threadIdx.x + o];
      s2[threadIdx.x] += s2[threadIdx.x + o];
    }
    __syncthreads();
  }
  if (threadIdx.x == 0) {
    ws[WS_STATS + blk * 2 + 0] = s1[0];
    ws[WS_STATS + blk * 2 + 1] = s2[0];
  }
  // zero the Gram accumulators (runs before gram kernel; stream-ordered)
  if (blk == 0 && threadIdx.x < B_ * 18) ws[WS_GRAM + threadIdx.x] = 0.f;
}

// ---------- kernel 2: G = Knorm · Vᵀ and G2 = Knorm · (V²)ᵀ via V_WMMA_F32_16X16X4_F32 ----------
// A tile (16x4): rows 0..2 = normalized F_s_prev channels, rest zero.
// B tile (4x16): cols 0..2 = F_s channels, cols 3..5 = F_s², rest zero.
// Accumulates K=4 s-values per WMMA; partial 3x6 results reduced with f32 global atomics.
__global__ void gram_wmma_kernel(const float* __restrict__ Fs,
                                 const float* __restrict__ Fsp,
                                 float* __restrict__ ws) {
  const int b     = blockIdx.x >> 3;     // 16 blocks: 8 per batch
  const int slice = blockIdx.x & 7;
  const int lane  = threadIdx.x & 31;
  const int wave  = threadIdx.x >> 5;    // 8 waves per block
  const int half  = lane >> 4;           // lane half selects K pair {0,1} vs {2,3}
  const int row   = lane & 15;           // A: M row ; B: N col

  // A-side params: channel row (valid rows 0..2), stats blocks 6..8 (F_s_prev)
  const int  ca    = (row < 2) ? row : 2;
  float meanA, rstdA;
  ch_stats(ws, 6 + ca, meanA, rstdA);
  const float amask = (row < C_) ? 1.0f : 0.0f;

  // B-side: col<3 -> V[col], 3<=col<6 -> V[col-3]^2
  const int   cb  = (row < C_) ? row : ((row < 6) ? (row - C_) : 2);
  const float sv  = (row < C_) ? 1.0f : 0.0f;
  const float sv2 = (row >= C_ && row < 6) ? 1.0f : 0.0f;

  const float* ap = Fsp + (size_t)(b * C_ + ca) * HW_;
  const float* bp = Fs  + (size_t)(b * C_ + cb) * HW_;

  v8f acc = {0.f, 0.f, 0.f, 0.f, 0.f, 0.f, 0.f, 0.f};

  // 9216/4 = 2304 K-chunks, strided over 8 blocks-per-batch-slices * 8 waves = 64 lanes of work
  for (int j = slice * 8 + wave; j < HW_ / 4; j += 64) {
    const int s = j * 4 + 2 * half;                          // even -> 8B-aligned b64 loads
    const v2f av = *reinterpret_cast<const v2f*>(ap + s);
    const v2f bv = *reinterpret_cast<const v2f*>(bp + s);

    v2f A;
    A.x = amask * (av.x - meanA) * rstdA;
    A.y = amask * (av.y - meanA) * rstdA;

    v2f Bv;
    Bv.x = sv * bv.x + sv2 * bv.x * bv.x;
    Bv.y = sv * bv.y + sv2 * bv.y * bv.y;

    // D = A(16x4) * B(4x16) + C, full f32  ->  v_wmma_f32_16x16x4_f32
    acc = __builtin_amdgcn_wmma_f32_16x16x4_f32(
        /*neg_a=*/false, A, /*neg_b=*/false, Bv,
        /*c_mod=*/(short)0, acc, /*reuse_a=*/false, /*reuse_b=*/false);
  }

  // D[m][n] (m<8, n<16) lives in VGPR m, lane n -> rows 0..2, cols 0..5 hold G|G2
  const float r0 = acc[0];
  const float r1 = acc[1];
  const float r2 = acc[2];
  if (lane < 6) {
    float* g = ws + WS_GRAM + b * 18;
    atomicAdd(&g[0 * 6 + lane], r0);
    atomicAdd(&g[1 * 6 + lane], r1);
    atomicAdd(&g[2 * 6 + lane], r2);
  }
}

// ---------- kernel 3: epilogue, one thread per (b, n) computing all 3 channels ----------
__global__ void out_kernel(const float* __restrict__ Fc,
                           const float* __restrict__ Fcp,
                           const float* __restrict__ ws,
                           float* __restrict__ out) {
  const int t = blockIdx.x * 256 + threadIdx.x;
  if (t >= B_ * HW_) return;
  const int b = t / HW_;
  const int n = t - b * HW_;

  float meanC[C_], rstdC[C_], meanQ[C_], rstdQ[C_];
#pragma unroll
  for (int c = 0; c < C_; ++c) {
    ch_stats(ws, 0 + c, meanC[c], rstdC[c]);   // F_c stats
    ch_stats(ws, 3 + c, meanQ[c], rstdQ[c]);   // F_c_prev stats
  }

  float Q[C_], Cn[C_];
#pragma unroll
  for (int c = 0; c < C_; ++c) {
    Q[c]  = (Fcp[(size_t)(b * C_ + c) * HW_ + n] - meanQ[c]) * rstdQ[c];
    Cn[c] = (Fc [(size_t)(b * C_ + c) * HW_ + n] - meanC[c]) * rstdC[c];
  }

  const float* g = ws + WS_GRAM + b * 18;
#pragma unroll
  for (int c = 0; c < C_; ++c) {
    const float M  = Q[0] * g[0 * 6 + c]     + Q[1] * g[1 * 6 + c]     + Q[2] * g[2 * 6 + c];
    const float M2 = Q[0] * g[0 * 6 + 3 + c] + Q[1] * g[1 * 6 + 3 + c] + Q[2] * g[2 * 6 + 3 + c];
    const float d  = M2 - M * M;
    const float S  = (d > 0.f) ? sqrtf(d) : 0.f;   // sqrt(neg)->NaN->0 per nan_to_num
    out[(size_t)(b * C_ + c) * HW_ + n] = S * Cn[c] + M;
  }
}

extern "C" void kernel_launch(void* const* d_in, const int* in_sizes, int n_in,
                              void* d_out, int out_size, void* d_ws, size_t ws_size,
                              hipStream_t stream) {
  (void)in_sizes; (void)n_in; (void)out_size; (void)ws_size;
  const float* Fc  = (const float*)d_in[0];  // F_c
  const float* Fs  = (const float*)d_in[1];  // F_s
  const float* Fcp = (const float*)d_in[2];  // F_c_previous
  const float* Fsp = (const float*)d_in[3];  // F_s_previous
  float* ws  = (float*)d_ws;
  float* out = (float*)d_out;

  stats_kernel<<<9, 256, 0, stream>>>(Fc, Fcp, Fsp, ws);
  gram_wmma_kernel<<<16, 256, 0, stream>>>(Fs, Fsp, ws);
  out_kernel<<<(B_ * HW_ + 255) / 256, 256, 0, stream>>>(Fc, Fcp, ws, out);
}